// MelSpectrogramLoss_86569360818594
// MI455X (gfx1250) — compile-verified
//
#include <hip/hip_runtime.h>
#include <math.h>

// ---------------------------------------------------------------------------
// MelSpectrogramLoss on gfx1250 (MI455X, wave32, WMMA)
//
// pipeline per wave (16 frames):
//   DFT-as-GEMM : A = raw frames (16 x 64, k=63 padded, window folded into B)
//                 B = [cos|sin] basis * hann  (64 x 64)   -> v_wmma_f32_16x16x4_f32
//   power       : P = cos^2 + sin^2          (elementwise, registers)
//   linearity   : Pdiff = P_pred - P_target  (mel is linear in power)
//   mel GEMM    : Pdiff (16 x 32) x FB (32 x 64)          -> v_wmma_f32_16x16x4_f32
//   loss        : sum |mel| (frames >= 4000 masked), deterministic reduction
//
// B/FB matrices are stored K-PAIR INTERLEAVED: element (k,j) at
//   (k>>1)*128 + j*2 + (k&1)
// so each WMMA B-fragment (two floats, K and K+1, same column) is one aligned
// ds_load_b64 straight into a contiguous VGPR pair (no v_mov shuffling).
// ---------------------------------------------------------------------------

typedef float v2f __attribute__((ext_vector_type(2)));
typedef float v8f __attribute__((ext_vector_type(8)));

#define T_LEN     64000
#define BATCH     64
#define N_FFT     63
#define HOP       16
#define PADW      31
#define N_FRAMES  4000          // 1 + (64000 + 62 - 63)/16
#define FR_PER_WG 64            // 4 waves x 16 frames
#define TILES_PB  63            // ceil(4000/64)
#define CHUNK     1072          // 63*16 + 63, rounded up (k=63 slot readable)
#define WGT       128

#define WMMA_F32(A, B, C) \
  __builtin_amdgcn_wmma_f32_16x16x4_f32(false, (A), false, (B), (short)0, (C), false, false)

static __device__ __forceinline__ v2f ld2(const float* p) {
  return *(const v2f*)p;   // 8-byte aligned by construction -> ds_load_b64
}

// d_ws layout (floats): [0,4096)    Bmat, k-pair interleaved (64x64, window folded)
//                       [4096,6144) FB,   k-pair interleaved (32x64)
//                       [6144,6144+4032) per-WG partial sums
// ---------------------------------------------------------------------------
__global__ void init_mats_kernel(float* __restrict__ mats) {
  int i = blockIdx.x * blockDim.x + threadIdx.x;
  const float TWO_PI = 6.28318530717958647692f;
  if (i < 4096) {
    int k = i >> 6, j = i & 63;            // logical (k, j)
    float v = 0.0f;
    if (k < 63) {
      float win = 0.5f * (1.0f - cosf(TWO_PI * (float)k / 63.0f));
      int prod = ((j & 31) * k) % 63;      // exact periodic reduction
      float ang = TWO_PI * (float)prod / 63.0f;
      v = ((j < 32) ? cosf(ang) : sinf(ang)) * win;  // sin sign irrelevant (squared)
    }
    mats[(k >> 1) * 128 + j * 2 + (k & 1)] = v;
  } else if (i < 6144) {
    int idx = i - 4096;
    int q = idx >> 6, m = idx & 63;        // logical (q, m)
    float freq = (float)q * (4000.0f / 31.0f);
    float mmax = 2595.0f * log10f(1.0f + 4000.0f / 700.0f);
    float f0 = 700.0f * (powf(10.0f, (mmax * (float)(m)     / 65.0f) / 2595.0f) - 1.0f);
    float f1 = 700.0f * (powf(10.0f, (mmax * (float)(m + 1) / 65.0f) / 2595.0f) - 1.0f);
    float f2 = 700.0f * (powf(10.0f, (mmax * (float)(m + 2) / 65.0f) / 2595.0f) - 1.0f);
    float down = (freq - f0) / (f1 - f0);
    float up   = (f2 - freq) / (f2 - f1);
    mats[4096 + (q >> 1) * 128 + m * 2 + (q & 1)] = fmaxf(0.0f, fminf(down, up));
  }
}

// ---------------------------------------------------------------------------
__global__ __launch_bounds__(WGT) void melspec_loss_kernel(
    const float* __restrict__ pred, const float* __restrict__ tgt,
    const float* __restrict__ mats, float* __restrict__ partials) {
  __shared__ __align__(16) float chunkP[CHUNK];
  __shared__ __align__(16) float chunkT[CHUNK];
  __shared__ __align__(16) float Bm[64 * 64];     // k-pair interleaved
  __shared__ __align__(16) float FBs[32 * 64];    // k-pair interleaved
  __shared__ __align__(16) float Ptr[4][16 * 32];
  __shared__ float red[WGT];

  const int tid  = threadIdx.x;
  const int b    = blockIdx.y;
  const int wgf0 = blockIdx.x * FR_PER_WG;
  const int p0   = wgf0 * HOP;

  // constant matrices: HBM/L2 -> LDS (shared by all WGs, L2-resident)
  for (int i = tid; i < 64 * 64; i += WGT) Bm[i]  = mats[i];
  for (int i = tid; i < 32 * 64; i += WGT) FBs[i] = mats[4096 + i];

  // signal chunks with reflect padding (each HBM sample reused by ~4 frames)
  const float* pb = pred + (size_t)b * T_LEN;
  const float* tb = tgt  + (size_t)b * T_LEN;
  for (int i = tid; i < CHUNK; i += WGT) {
    int org = p0 + i - PADW;
    org = (org < 0) ? -org : org;
    org = (org >= T_LEN) ? (2 * (T_LEN - 1) - org) : org;
    chunkP[i] = pb[org];
    chunkT[i] = tb[org];
  }
  __syncthreads();

  const int lane  = tid & 31;
  const int wid   = tid >> 5;
  const int n     = lane & 15;      // tile column (also A-row for GEMMs)
  const int half  = lane >> 4;      // wave half
  const int kb    = half * 2;       // K base within K=4 group (ISA A/B layout)
  const int mb    = half * 8;       // C/D row base (ISA 16x16 f32 layout)
  const int fbase = wid * 16;       // this wave's frame block (local)

  // ---- stage 1: DFT GEMM + power, both inputs --------------------------------
  v8f Ppow[2][2];
  for (int inp = 0; inp < 2; ++inp) {
    const float* ch = inp ? chunkT : chunkP;
    v8f a0 = {}, a1 = {}, a2 = {}, a3 = {};
    #pragma unroll
    for (int ks = 0; ks < 16; ++ks) {
      const int k = ks * 4 + kb;                 // even; k=63 col hits Bm pair == 0
      v2f a = ld2(&ch[(fbase + n) * HOP + k]);   // contiguous, 8B aligned
      const float* bp = &Bm[(k >> 1) * 128];     // this lane-half's K pair
      v2f b0 = ld2(&bp[n * 2]);                  // (K,K+1) x col j  -> one b64
      v2f b1 = ld2(&bp[(16 + n) * 2]);
      v2f b2 = ld2(&bp[(32 + n) * 2]);
      v2f b3 = ld2(&bp[(48 + n) * 2]);
      a0 = WMMA_F32(a, b0, a0);                  // cos, q = 0..15
      a1 = WMMA_F32(a, b1, a1);                  // cos, q = 16..31
      a2 = WMMA_F32(a, b2, a2);                  // sin, q = 0..15
      a3 = WMMA_F32(a, b3, a3);                  // sin, q = 16..31
    }
    v8f q0, q1;
    #pragma unroll
    for (int v = 0; v < 8; ++v) {
      q0[v] = a0[v] * a0[v] + a2[v] * a2[v];     // |X|^2, q = 0..15
      q1[v] = a1[v] * a1[v] + a3[v] * a3[v];     // |X|^2, q = 16..31
    }
    Ppow[inp][0] = q0;
    Ppow[inp][1] = q1;
  }

  // ---- stage 2: Pdiff transpose through LDS (D-layout -> A-layout) -----------
  float* pw = Ptr[wid];
  #pragma unroll
  for (int v = 0; v < 8; ++v) {
    pw[(mb + v) * 32 + n]      = Ppow[0][0][v] - Ppow[1][0][v];
    pw[(mb + v) * 32 + 16 + n] = Ppow[0][1][v] - Ppow[1][1][v];
  }
  __syncthreads();

  // ---- stage 3: mel GEMM on the power difference -----------------------------
  v8f m0 = {}, m1 = {}, m2 = {}, m3 = {};
  #pragma unroll
  for (int ks = 0; ks < 8; ++ks) {
    const int k = ks * 4 + kb;                   // even
    v2f a = ld2(&pw[n * 32 + k]);
    const float* fp = &FBs[(k >> 1) * 128];
    v2f b0 = ld2(&fp[n * 2]);
    v2f b1 = ld2(&fp[(16 + n) * 2]);
    v2f b2 = ld2(&fp[(32 + n) * 2]);
    v2f b3 = ld2(&fp[(48 + n) * 2]);
    m0 = WMMA_F32(a, b0, m0);
    m1 = WMMA_F32(a, b1, m1);
    m2 = WMMA_F32(a, b2, m2);
    m3 = WMMA_F32(a, b3, m3);
  }

  // ---- stage 4: masked |.| sum, deterministic WG tree reduction --------------
  float s = 0.0f;
  #pragma unroll
  for (int v = 0; v < 8; ++v) {
    int fg = wgf0 + fbase + mb + v;
    if (fg < N_FRAMES)
      s += fabsf(m0[v]) + fabsf(m1[v]) + fabsf(m2[v]) + fabsf(m3[v]);
  }
  red[tid] = s;
  __syncthreads();
  for (int off = WGT / 2; off > 0; off >>= 1) {
    if (tid < off) red[tid] += red[tid + off];
    __syncthreads();
  }
  if (tid == 0) partials[b * TILES_PB + blockIdx.x] = red[0];
}

// ---------------------------------------------------------------------------
__global__ void finalize_kernel(const float* __restrict__ partials,
                                float* __restrict__ out) {
  __shared__ float red[256];
  float s = 0.0f;
  for (int i = threadIdx.x; i < BATCH * TILES_PB; i += 256) s += partials[i];
  red[threadIdx.x] = s;
  __syncthreads();
  for (int off = 128; off > 0; off >>= 1) {
    if (threadIdx.x < off) red[threadIdx.x] += red[threadIdx.x + off];
    __syncthreads();
  }
  if (threadIdx.x == 0)
    out[0] = red[0] * (1.0f / ((float)BATCH * 64.0f * (float)N_FRAMES));
}

// ---------------------------------------------------------------------------
extern "C" void kernel_launch(void* const* d_in, const int* in_sizes, int n_in,
                              void* d_out, int out_size, void* d_ws, size_t ws_size,
                              hipStream_t stream) {
  (void)in_sizes; (void)n_in; (void)out_size; (void)ws_size;
  const float* pred = (const float*)d_in[0];
  const float* tgt  = (const float*)d_in[1];
  float* ws       = (float*)d_ws;
  float* mats     = ws;             // 6144 floats
  float* partials = ws + 6144;      // 4032 floats

  init_mats_kernel<<<(6144 + 255) / 256, 256, 0, stream>>>(mats);
  dim3 grid(TILES_PB, BATCH);
  melspec_loss_kernel<<<grid, WGT, 0, stream>>>(pred, tgt, mats, partials);
  finalize_kernel<<<1, 256, 0, stream>>>(partials, (float*)d_out);
}